// KMeansLayer_28501402977026
// MI455X (gfx1250) — compile-verified
//
#include <hip/hip_runtime.h>
#include <math.h>

typedef __attribute__((ext_vector_type(2))) float v2f;
typedef __attribute__((ext_vector_type(8))) float v8f;

#define K_CLUSTERS 512
#define DIM 64
#define TEMP_INV 10.0f          // 1/T, T = 0.1
#define ROWS_PER_WG 16
#define WAVES_PER_WG 4
#define COLS_PER_WAVE 128
#define TILES_PER_WAVE 8        // 8 tiles * 16 cols = 128 cols per wave

// ---------------------------------------------------------------------------
// Pre-kernel: csqT[n] = (sum_d centroids[n,d]^2) / T
// ---------------------------------------------------------------------------
__global__ void kmeans_csq_kernel(const float* __restrict__ cent,
                                  float* __restrict__ csqT) {
    int n = blockIdx.x * blockDim.x + threadIdx.x;
    if (n < K_CLUSTERS) {
        const float4* row = (const float4*)(cent + (size_t)n * DIM);
        float s = 0.0f;
#pragma unroll
        for (int i = 0; i < DIM / 4; ++i) {
            float4 v = row[i];
            s += v.x * v.x + v.y * v.y + v.z * v.z + v.w * v.w;
        }
        csqT[n] = s * TEMP_INV;
    }
}

// ---------------------------------------------------------------------------
// Main kernel: for a block of 16 rows, compute logits = (2*x.c - c^2)/T via
// fp32 WMMA (16x16x4), then fused softmax over the 512 clusters.
// ---------------------------------------------------------------------------
__global__ __launch_bounds__(128) void kmeans_softmax_kernel(
    const float* __restrict__ x, const float* __restrict__ cent,
    const float* __restrict__ csqT, float* __restrict__ out) {

    __shared__ float xtile[ROWS_PER_WG * DIM];            // 4 KB
    __shared__ float red[WAVES_PER_WG][ROWS_PER_WG];      // partial max / sum
    __shared__ float gred[ROWS_PER_WG];                   // global max / 1/sum

    const int tid  = threadIdx.x;
    const int wave = tid >> 5;
    const int lane = tid & 31;
    const int half = lane >> 4;    // 0: lanes 0-15, 1: lanes 16-31
    const int l16  = lane & 15;
    const size_t rowbase = (size_t)blockIdx.x * ROWS_PER_WG;

    // ---- stage x tile (16 rows x 64) into LDS, coalesced float4 ----
    {
        const float4* src = (const float4*)(x + rowbase * DIM);
        float4* dst = (float4*)xtile;
        dst[tid]       = src[tid];
        dst[tid + 128] = src[tid + 128];
    }
    __syncthreads();

    // ---- preload A fragments: A[M = l16, K = 4*kc + 2*half + {0,1}] ----
    v2f afrag[16];
#pragma unroll
    for (int kc = 0; kc < 16; ++kc) {
        afrag[kc] = *(const v2f*)&xtile[l16 * DIM + 4 * kc + 2 * half];
    }

    // ---- WMMA: 8 column tiles of 16, K=64 as 16 chained 16x16x4 steps ----
    const int wavecol = wave * COLS_PER_WAVE;
    v8f acc[TILES_PER_WAVE];
#pragma unroll
    for (int t = 0; t < TILES_PER_WAVE; ++t) {
        const int col = wavecol + t * 16 + l16;
        const float* bbase = cent + (size_t)col * DIM + 2 * half;
        v8f c = {};
#pragma unroll
        for (int kc = 0; kc < 16; ++kc) {
            v2f b = *(const v2f*)(bbase + 4 * kc);
            c = __builtin_amdgcn_wmma_f32_16x16x4_f32(
                    false, afrag[kc], false, b, (short)0, c, false, false);
        }
        acc[t] = c;
    }

    // ---- logits = 2/T * cross - c^2/T  (||x||^2 cancels in softmax) ----
    float csq[TILES_PER_WAVE];
#pragma unroll
    for (int t = 0; t < TILES_PER_WAVE; ++t)
        csq[t] = csqT[wavecol + t * 16 + l16];
#pragma unroll
    for (int t = 0; t < TILES_PER_WAVE; ++t)
#pragma unroll
        for (int v = 0; v < 8; ++v)
            acc[t][v] = acc[t][v] * (2.0f * TEMP_INV) - csq[t];

    // C/D layout: VGPR v, lane group `half` holds row r = v + 8*half,
    // column = tile_col_base + l16.

    // ---- pass 1: per-row max ----
    float rmax[8];
#pragma unroll
    for (int v = 0; v < 8; ++v) {
        float m = acc[0][v];
#pragma unroll
        for (int t = 1; t < TILES_PER_WAVE; ++t) m = fmaxf(m, acc[t][v]);
#pragma unroll
        for (int mask = 1; mask < 16; mask <<= 1)
            m = fmaxf(m, __shfl_xor(m, mask, 32));
        rmax[v] = m;
    }
    if (l16 == 0) {
#pragma unroll
        for (int v = 0; v < 8; ++v) red[wave][v + 8 * half] = rmax[v];
    }
    __syncthreads();
    if (tid < ROWS_PER_WG) {
        float m = red[0][tid];
#pragma unroll
        for (int w = 1; w < WAVES_PER_WG; ++w) m = fmaxf(m, red[w][tid]);
        gred[tid] = m;
    }
    __syncthreads();
    float gmax[8];
#pragma unroll
    for (int v = 0; v < 8; ++v) gmax[v] = gred[v + 8 * half];

    // ---- pass 2: exponentiate in place, per-row sum ----
    float rsum[8];
#pragma unroll
    for (int v = 0; v < 8; ++v) rsum[v] = 0.0f;
#pragma unroll
    for (int t = 0; t < TILES_PER_WAVE; ++t)
#pragma unroll
        for (int v = 0; v < 8; ++v) {
            float e = __expf(acc[t][v] - gmax[v]);
            acc[t][v] = e;
            rsum[v] += e;
        }
#pragma unroll
    for (int v = 0; v < 8; ++v) {
#pragma unroll
        for (int mask = 1; mask < 16; mask <<= 1)
            rsum[v] += __shfl_xor(rsum[v], mask, 32);
    }
    if (l16 == 0) {
#pragma unroll
        for (int v = 0; v < 8; ++v) red[wave][v + 8 * half] = rsum[v];
    }
    __syncthreads();
    if (tid < ROWS_PER_WG) {
        float s = red[0][tid];
#pragma unroll
        for (int w = 1; w < WAVES_PER_WG; ++w) s += red[w][tid];
        gred[tid] = 1.0f / s;
    }
    __syncthreads();
    float rinv[8];
#pragma unroll
    for (int v = 0; v < 8; ++v) rinv[v] = gred[v + 8 * half];

    // ---- write normalized assignments ----
#pragma unroll
    for (int t = 0; t < TILES_PER_WAVE; ++t) {
        const int col = wavecol + t * 16 + l16;
#pragma unroll
        for (int v = 0; v < 8; ++v) {
            const size_t r = rowbase + (size_t)(v + 8 * half);
            out[r * K_CLUSTERS + col] = acc[t][v] * rinv[v];
        }
    }
}

// ---------------------------------------------------------------------------
extern "C" void kernel_launch(void* const* d_in, const int* in_sizes, int n_in,
                              void* d_out, int out_size, void* d_ws, size_t ws_size,
                              hipStream_t stream) {
    const float* x    = (const float*)d_in[0];   // (8, 32768, 64) f32
    const float* cent = (const float*)d_in[1];   // (512, 64) f32
    float* out  = (float*)d_out;                 // (8, 32768, 512) f32
    float* csqT = (float*)d_ws;                  // 512 floats of scratch

    const long nrows = (long)in_sizes[0] / DIM;  // 262144

    kmeans_csq_kernel<<<(K_CLUSTERS + 255) / 256, 256, 0, stream>>>(cent, csqT);

    const int nblocks = (int)(nrows / ROWS_PER_WG);   // 16384 workgroups
    kmeans_softmax_kernel<<<nblocks, 128, 0, stream>>>(x, cent, csqT, out);
}